// GConvLayer_15333033247245
// MI455X (gfx1250) — compile-verified
//
#include <hip/hip_runtime.h>
#include <hip/hip_bf16.h>

typedef __attribute__((ext_vector_type(2))) float v2f;
typedef __attribute__((ext_vector_type(8))) float v8f;

#define HIDDEN 64
#define EPS 1e-5f

// ---------------------------------------------------------------------------
// Kernel 1: zero the aggregation buffer (d_out doubles as agg accumulator).
// ---------------------------------------------------------------------------
__global__ __launch_bounds__(256) void k_zero(float4* __restrict__ p, int n4) {
    int t = blockIdx.x * blockDim.x + threadIdx.x;
    if (t < n4) p[t] = make_float4(0.f, 0.f, 0.f, 0.f);
}

// ---------------------------------------------------------------------------
// Kernel 2: m = relu(x @ W^T + b) via V_WMMA_F32_16X16X4_F32.
// One wave computes one 16x16 tile of m; K-dim (=64) consumed 4 at a time.
//
// f32 WMMA operand layouts (ISA 7.12.2):
//   A (16x4, MxK):  lane L<16 -> row M=L,    regs {K=0, K=1}
//                   lane L>=16 -> row M=L-16, regs {K=2, K=3}
//   B (4x16, KxN):  lane L<16 -> col N=L,    regs {K=0, K=1}
//                   lane L>=16 -> col N=L-16, regs {K=2, K=3}
//   C/D (16x16):    reg r, lane L -> element (M = r + 8*(L>=16), N = L&15)
// ---------------------------------------------------------------------------
__global__ __launch_bounds__(256) void k_gemm_relu(
    const float* __restrict__ x, const float* __restrict__ W,
    const float* __restrict__ bias, float* __restrict__ m, int n_nodes) {
    int wave = (blockIdx.x * blockDim.x + threadIdx.x) >> 5;
    int lane = threadIdx.x & 31;
    int rowTile = wave >> 2;   // node tile (16 nodes)
    int colTile = wave & 3;    // output-feature tile (16 feats)
    if (rowTile * 16 >= n_nodes) return;   // wave-uniform guard

    int row0 = rowTile * 16;
    int col0 = colTile * 16;
    int half = lane >> 4;      // 0: K pair {0,1}; 1: K pair {2,3}
    int l    = lane & 15;

    const float* xrow = x + (size_t)(row0 + l) * HIDDEN;  // A row for this lane
    const float* wrow = W + (size_t)(col0 + l) * HIDDEN;  // B col == W row (W[d,k])

    v8f c = {};
#pragma unroll
    for (int k0 = 0; k0 < HIDDEN; k0 += 4) {
        int ka = k0 + half * 2;
        v2f a = *(const v2f*)(xrow + ka);
        v2f b = *(const v2f*)(wrow + ka);
        c = __builtin_amdgcn_wmma_f32_16x16x4_f32(
                /*neg_a=*/false, a, /*neg_b=*/false, b,
                /*c_mod=*/(short)0, c, /*reuse_a=*/false, /*reuse_b=*/false);
    }

    float bv = bias[col0 + l];
#pragma unroll
    for (int r = 0; r < 8; ++r) {
        float v = c[r] + bv;
        v = v > 0.f ? v : 0.f;
        m[(size_t)(row0 + r + half * 8) * HIDDEN + col0 + l] = v;
    }
}

// ---------------------------------------------------------------------------
// Kernel 3: agg[dst] += m[src]   (16 threads per edge, float4 each,
// hardware global_atomic_add_f32 via unsafeAtomicAdd; L2-resident).
// ---------------------------------------------------------------------------
__global__ __launch_bounds__(256) void k_scatter_add(
    const float* __restrict__ m, const int* __restrict__ dst,
    const int* __restrict__ src, float* __restrict__ agg, int n_edges) {
    int t = blockIdx.x * blockDim.x + threadIdx.x;
    int e = t >> 4;
    if (e >= n_edges) return;
    int f = (t & 15) * 4;
    int s = src[e];
    int d = dst[e];
    float4 v = *(const float4*)(m + (size_t)s * HIDDEN + f);
    float* p = agg + (size_t)d * HIDDEN + f;
    unsafeAtomicAdd(p + 0, v.x);
    unsafeAtomicAdd(p + 1, v.y);
    unsafeAtomicAdd(p + 2, v.z);
    unsafeAtomicAdd(p + 3, v.w);
}

// ---------------------------------------------------------------------------
// Kernel 4: h = x + agg; out = nw * h * rsqrt(mean(h^2)+eps) + nb.
// One wave32 per node, float2 per lane, butterfly reduction.
// agg is read from `out` and overwritten in place.
// ---------------------------------------------------------------------------
__global__ __launch_bounds__(256) void k_residual_rmsnorm(
    const float* __restrict__ x, const float* __restrict__ nw,
    const float* __restrict__ nb, float* __restrict__ out, int n_nodes) {
    int wave = (blockIdx.x * blockDim.x + threadIdx.x) >> 5;
    int lane = threadIdx.x & 31;
    if (wave >= n_nodes) return;   // wave-uniform guard

    size_t base = (size_t)wave * HIDDEN + lane * 2;
    float2 xv = *(const float2*)(x + base);
    float2 av = *(const float2*)(out + base);
    float h0 = xv.x + av.x;
    float h1 = xv.y + av.y;

    float ss = h0 * h0 + h1 * h1;
#pragma unroll
    for (int mask = 16; mask >= 1; mask >>= 1)
        ss += __shfl_xor(ss, mask, 32);

    float inv = rsqrtf(ss * (1.0f / HIDDEN) + EPS);
    float w0 = nw[lane * 2 + 0], w1 = nw[lane * 2 + 1];
    float b0 = nb[lane * 2 + 0], b1 = nb[lane * 2 + 1];
    float2 o;
    o.x = w0 * (h0 * inv) + b0;
    o.y = w1 * (h1 * inv) + b1;
    *(float2*)(out + base) = o;
}

// ---------------------------------------------------------------------------
extern "C" void kernel_launch(void* const* d_in, const int* in_sizes, int n_in,
                              void* d_out, int out_size, void* d_ws, size_t ws_size,
                              hipStream_t stream) {
    const float* x     = (const float*)d_in[0];
    const int*   edges = (const int*)  d_in[1];
    const float* W     = (const float*)d_in[2];
    const float* b     = (const float*)d_in[3];
    const float* nw    = (const float*)d_in[4];
    const float* nb    = (const float*)d_in[5];
    float* out = (float*)d_out;
    float* m   = (float*)d_ws;   // [n_nodes, 64] message buffer (25.6 MB)

    int n_nodes = in_sizes[0] / HIDDEN;
    int n_edges = in_sizes[1] / 2;
    const int* dst = edges;            // edges[0,:]
    const int* src = edges + n_edges;  // edges[1,:]

    // 1) zero agg (d_out)
    int n4 = (n_nodes * HIDDEN) / 4;
    k_zero<<<(n4 + 255) / 256, 256, 0, stream>>>((float4*)out, n4);

    // 2) m = relu(x @ W^T + b) : 16x16 tile per wave, 8 waves per block
    int nwaves = ((n_nodes + 15) / 16) * 4;
    int nblk_g = (nwaves * 32 + 255) / 256;
    k_gemm_relu<<<nblk_g, 256, 0, stream>>>(x, W, b, m, n_nodes);

    // 3) scatter-add messages along edges
    long long nthr = (long long)n_edges * 16;
    int nblk_s = (int)((nthr + 255) / 256);
    k_scatter_add<<<nblk_s, 256, 0, stream>>>(m, dst, src, out, n_edges);

    // 4) residual + RMSNorm, one wave per node
    int nblk_n = (n_nodes * 32 + 255) / 256;
    k_residual_rmsnorm<<<nblk_n, 256, 0, stream>>>(x, nw, nb, out, n_nodes);
}